// Sobel_operation_c_52604759441690
// MI455X (gfx1250) — compile-verified
//
#include <hip/hip_runtime.h>
#include <math.h>
#include <stdint.h>

// ---------------------------------------------------------------------------
// Sobel magnitude, B=16 C=64 H=256 W=256 f32, zero padding.
// Memory-bound (>=512MB traffic @ 23.3TB/s ~ 22us floor) -> optimize data path:
// CDNA5 Tensor Data Mover DMA of a (TILE+2) x W f32 tile into LDS, separable
// Sobel compute from LDS, 128-bit coalesced stores. Raw v_sqrt_f32 for the
// magnitude (no libm denorm fixup tail).
// ---------------------------------------------------------------------------

typedef unsigned int v4u __attribute__((ext_vector_type(4)));
typedef int          v4i __attribute__((ext_vector_type(4)));
typedef int          v8i __attribute__((ext_vector_type(8)));

#define SOB_B   16
#define SOB_C   64
#define SOB_H   256
#define SOB_W   256
#define TILE    32
#define TILES_PER_IMAGE (SOB_H / TILE)     // 8 (power of two -> shift/mask)
#define LDS_ROWS (TILE + 2)

__global__ __launch_bounds__(256) void sobel_tdm_kernel(const float* __restrict__ x,
                                                        float* __restrict__ out) {
    __shared__ float smem[LDS_ROWS * SOB_W];   // 34 KB

    const int tile = blockIdx.x;
    const int bc   = tile >> 3;                // which (b,c) image plane
    const int r0   = (tile & (TILES_PER_IMAGE - 1)) * TILE;   // first output row

    // Input rows needed: [r0-1, r0+TILE]. LDS row i holds input row (r0-1+i).
    const int gstart      = (r0 == 0) ? 0 : (r0 - 1);
    const int ldsRowStart = (r0 == 0) ? 1 : 0;
    const int nrows       = LDS_ROWS - ldsRowStart;

    // Top zero-pad row (only for the first tile of each image).
    if (r0 == 0 && threadIdx.x < 64) {
        float4 z = make_float4(0.f, 0.f, 0.f, 0.f);
        *(float4*)&smem[threadIdx.x * 4] = z;
    }

    // Wave 0 issues one TDM load of the whole tile; bottom zero-padding comes
    // from TDM out-of-bounds reads returning zero (tensor_dim1 = H - gstart).
    // Waves 1..7 skip via s_cbranch_execz (TDM ignores EXEC, so the branch is
    // what guarantees exactly one DMA per workgroup).
    if (threadIdx.x < 32) {
        const unsigned long long ga =
            (unsigned long long)(uintptr_t)(x + ((size_t)bc * SOB_H + (size_t)gstart) * SOB_W);
        const unsigned int lds_base =
            (unsigned int)(uintptr_t)(&smem[ldsRowStart * SOB_W]);

        // ---- D# group 0 (4 dwords) ----
        v4u g0;
        g0.x = 1u;                                        // count=1, user descriptor
        g0.y = lds_base;                                  // lds_addr (bytes)
        g0.z = (unsigned int)(ga & 0xFFFFFFFFull);        // global_addr[31:0]
        g0.w = (unsigned int)(ga >> 32) | (2u << 30);     // global_addr[56:32] | type=2

        // ---- D# group 1 (8 dwords) ----
        const unsigned int tdim0  = SOB_W;                      // tensor dim0 (elements)
        const unsigned int tdim1  = (unsigned int)(SOB_H - gstart); // rows >= this read as 0
        const unsigned int tile0  = SOB_W;                      // tile dim0
        const unsigned int tile1  = (unsigned int)nrows;        // tile dim1 (rows)
        const unsigned long long stride0 = SOB_W;               // row stride (elements)

        v8i g1;
        g1[0] = (int)(2u << 16);                                // data_size=2 (4B), mask=0
        g1[1] = (int)((tdim0 & 0xFFFFu) << 16);                 // tensor_dim0[15:0]
        g1[2] = (int)(((tdim0 >> 16) & 0xFFFFu) | ((tdim1 & 0xFFFFu) << 16));
        g1[3] = (int)(((tdim1 >> 16) & 0xFFFFu) | (tile0 << 16));
        g1[4] = (int)(tile1 & 0xFFFFu);                         // tile_dim1 | tile_dim2=0
        g1[5] = (int)(unsigned int)(stride0 & 0xFFFFFFFFull);   // dim0_stride[31:0]
        g1[6] = (int)(unsigned int)((stride0 >> 32) & 0xFFFFu); // dim0_stride[47:32]
        g1[7] = 0;

        v4i g2 = {0, 0, 0, 0};   // 2D tile: groups 2/3 unused
        v4i g3 = {0, 0, 0, 0};

#if defined(__clang_major__) && (__clang_major__ >= 23)
        v8i gz = {0, 0, 0, 0, 0, 0, 0, 0};
        __builtin_amdgcn_tensor_load_to_lds(g0, g1, g2, g3, gz, 0);
#else
        __builtin_amdgcn_tensor_load_to_lds(g0, g1, g2, g3, 0);
#endif
        __builtin_amdgcn_s_wait_tensorcnt(0);
    }
    __syncthreads();

    // Compute: each thread owns 4 columns (float4), 64 threads span a row,
    // 256 threads cover 4 rows per iteration, 8 unrolled iterations = 32 rows.
    const int cg = (threadIdx.x & 63) * 4;     // first column of this thread
    const int ry = threadIdx.x >> 6;           // row phase 0..3
    float* obase = out + (size_t)bc * SOB_H * SOB_W + (size_t)r0 * SOB_W + cg;

#pragma unroll
    for (int it = 0; it < TILE / 4; ++it) {
        const int rr = ry + it * 4;
        // LDS rows rr, rr+1, rr+2 hold input rows (r0+rr-1 .. r0+rr+1).
        float A[3][4];   // horizontal smooth [1,2,1]
        float D[3][4];   // horizontal diff   [1,0,-1] (left - right)
#pragma unroll
        for (int j = 0; j < 3; ++j) {
            const float* row = &smem[(rr + j) * SOB_W];
            float4 m = *(const float4*)(row + cg);
            float l = (cg == 0)          ? 0.f : row[cg - 1];
            float r = (cg == SOB_W - 4)  ? 0.f : row[cg + 4];
            A[j][0] = fmaf(2.f, m.x, l + m.y);
            A[j][1] = fmaf(2.f, m.y, m.x + m.z);
            A[j][2] = fmaf(2.f, m.z, m.y + m.w);
            A[j][3] = fmaf(2.f, m.w, m.z + r);
            D[j][0] = l   - m.y;
            D[j][1] = m.x - m.z;
            D[j][2] = m.y - m.w;
            D[j][3] = m.z - r;
        }
        float4 o;
        float gx, gy;
        gx = fmaf(2.f, D[1][0], D[0][0] + D[2][0]); gy = A[0][0] - A[2][0];
        o.x = __builtin_amdgcn_sqrtf(fmaf(gx, gx, gy * gy));
        gx = fmaf(2.f, D[1][1], D[0][1] + D[2][1]); gy = A[0][1] - A[2][1];
        o.y = __builtin_amdgcn_sqrtf(fmaf(gx, gx, gy * gy));
        gx = fmaf(2.f, D[1][2], D[0][2] + D[2][2]); gy = A[0][2] - A[2][2];
        o.z = __builtin_amdgcn_sqrtf(fmaf(gx, gx, gy * gy));
        gx = fmaf(2.f, D[1][3], D[0][3] + D[2][3]); gy = A[0][3] - A[2][3];
        o.w = __builtin_amdgcn_sqrtf(fmaf(gx, gx, gy * gy));

        *(float4*)(obase + (size_t)rr * SOB_W) = o;
    }
}

extern "C" void kernel_launch(void* const* d_in, const int* in_sizes, int n_in,
                              void* d_out, int out_size, void* d_ws, size_t ws_size,
                              hipStream_t stream) {
    (void)in_sizes; (void)n_in; (void)d_ws; (void)ws_size; (void)out_size;
    const float* x = (const float*)d_in[0];   // [16,64,256,256] f32
    // d_in[1], d_in[2] are the fixed Sobel kernels; coefficients are baked in.
    float* out = (float*)d_out;

    const int blocks = SOB_B * SOB_C * TILES_PER_IMAGE;     // 8192
    sobel_tdm_kernel<<<blocks, 256, 0, stream>>>(x, out);
}